// SortModel_71614284693830
// MI455X (gfx1250) — compile-verified
//
#include <hip/hip_runtime.h>
#include <stdint.h>

// ---------------------------------------------------------------------------
// SortModel gap-loss on MI455X (gfx1250):
//   clamp -> stable 4x8-bit LSD radix sort of (x,y) pairs ->
//   bucket-LUT-accelerated searchsorted/interp -> deterministic reduction.
// CDNA5 paths: TDM tensor_load_to_lds tile staging (+ s_wait_tensorcnt),
// v_wmma_f32_16x16x4_f32 for the 256-bin scan, wave32 ballot ranking.
// ---------------------------------------------------------------------------

#define SORT_TILE   8192
#define SORT_BLOCK  256
#define NWAVES      8
#define WAVE_CHUNK  (SORT_TILE / NWAVES)   // 1024
#define WAVE_ITEMS  (WAVE_CHUNK / 32)      // 32
#define LUT_SHIFT   14
#define LUT_N       65040                  // > (0x3F800000>>14)+2
#define ALPHA_F     10.0f
#define DELTA_F     0.0005f

typedef unsigned int v4u __attribute__((ext_vector_type(4)));
typedef int          v4i __attribute__((ext_vector_type(4)));
typedef int          v8i __attribute__((ext_vector_type(8)));
typedef float        v2f __attribute__((ext_vector_type(2)));
typedef float        v8f __attribute__((ext_vector_type(8)));

// ----------------------------- TDM 1-D tile load ---------------------------
// Copies tile_elems u32 words from gsrc into LDS at lds_byte_off.
// tensor_dim0 = avail_elems => hardware zero-fills reads past the buffer end.
__device__ __forceinline__ void tdm_load_1d(uint32_t lds_byte_off, const void* gsrc,
                                            uint32_t tile_elems, uint32_t avail_elems) {
  uint64_t ga = (uint64_t)(uintptr_t)gsrc;
  v4u g0;
  g0.x = 1u;                                   // count=1 valid descriptor
  g0.y = lds_byte_off;                         // lds_addr (bytes)
  g0.z = (uint32_t)ga;                         // global_addr[31:0]
  g0.w = (uint32_t)((ga >> 32) & 0x01FFFFFFu) | (2u << 30);  // addr[56:32] | type=2
  v8i g1;
  g1[0] = (int)(2u << 16);                     // wg_mask=0, data_size=2 (4 bytes)
  g1[1] = (int)((avail_elems & 0xFFFFu) << 16);        // tensor_dim0[15:0]
  g1[2] = (int)((avail_elems >> 16) & 0xFFFFu);        // tensor_dim0[31:16], dim1=0
  g1[3] = (int)((tile_elems & 0xFFFFu) << 16);         // tile_dim0
  g1[4] = 0;                                   // tile_dim1=0, tile_dim2=0 (unused)
  g1[5] = (int)tile_elems;                     // tensor_dim0_stride (benign)
  g1[6] = 0;
  g1[7] = 0;
  v4i g2 = {1, 0, 0, 0};                       // tensor_dim2=1
  v4i g3 = {0, 0, 0, 0};
#if __clang_major__ >= 23
  v8i g1b = {0, 0, 0, 0, 0, 0, 0, 0};
  __builtin_amdgcn_tensor_load_to_lds(g0, g1, g2, g3, g1b, 0);
#else
  __builtin_amdgcn_tensor_load_to_lds(g0, g1, g2, g3, 0);
#endif
}

// ------------------------ 256-wide exclusive scan (LDS) --------------------
__device__ __forceinline__ uint32_t block_scan_excl_256(uint32_t v, uint32_t* s,
                                                        uint32_t* total) {
  int t = threadIdx.x;
  uint32_t acc = v;
  s[t] = acc;
  __syncthreads();
#pragma unroll
  for (int off = 1; off < 256; off <<= 1) {
    uint32_t y = (t >= off) ? s[t - off] : 0u;
    __syncthreads();
    acc += y;
    s[t] = acc;
    __syncthreads();
  }
  if (total) *total = s[255];
  return acc - v;
}

// ------------------------------- kernels -----------------------------------

__global__ __launch_bounds__(256) void prep_kernel(const float* __restrict__ idx,
                                                   const float* __restrict__ arr,
                                                   uint32_t* __restrict__ keys,
                                                   uint32_t* __restrict__ vals, int n) {
  int i = blockIdx.x * 256 + threadIdx.x;
  if (i < n) {
    float c = fminf(fmaxf(idx[i], 0.0f), 1.0f);
    uint32_t k = __float_as_uint(c);
    if (k == 0x80000000u) k = 0u;  // -0.0 -> +0.0 so uint order == float order
    keys[i] = k;
    vals[i] = __float_as_uint(arr[i]);
  }
}

__global__ __launch_bounds__(256) void hist_kernel(const uint32_t* __restrict__ srcK,
                                                   uint32_t* __restrict__ hist,
                                                   int n, int numTiles, int shift) {
  __shared__ __align__(16) uint32_t ldsK[SORT_TILE];
  __shared__ uint32_t cnt[256];
  int tile = blockIdx.x, t = threadIdx.x;
  int tileStart = tile * SORT_TILE;
  int vcount = n - tileStart; if (vcount > SORT_TILE) vcount = SORT_TILE;
  cnt[t] = 0u;
  if (t < 32) {  // wave 0 issues the TDM copy, waits, then joins the barrier
    tdm_load_1d((uint32_t)(uintptr_t)(void*)ldsK, srcK + tileStart,
                SORT_TILE, (uint32_t)(n - tileStart));
    __builtin_amdgcn_s_wait_tensorcnt(0);
  }
  __syncthreads();
  for (int i = t; i < vcount; i += 256)
    atomicAdd(&cnt[(ldsK[i] >> shift) & 255u], 1u);
  __syncthreads();
  hist[(size_t)t * (size_t)numTiles + (size_t)tile] = cnt[t];
}

__global__ __launch_bounds__(256) void scanbins_kernel(uint32_t* __restrict__ hist,
                                                       uint32_t* __restrict__ binTotals,
                                                       int numTiles) {
  __shared__ uint32_t sbuf[256];
  int bin = blockIdx.x, t = threadIdx.x;
  uint32_t* row = hist + (size_t)bin * (size_t)numTiles;
  int per = (numTiles + 255) / 256;
  int lo = t * per, hi = lo + per; if (hi > numTiles) hi = numTiles;
  uint32_t s = 0u;
  for (int i = lo; i < hi; ++i) s += row[i];
  uint32_t total;
  uint32_t base = block_scan_excl_256(s, sbuf, &total);
  uint32_t run = base;
  for (int i = lo; i < hi; ++i) { uint32_t v = row[i]; row[i] = run; run += v; }
  if (t == 255) binTotals[bin] = total;
}

// Exclusive scan of 256 bin totals via WMMA: treat totals as 16x16 A, compute
// D = A*U + C  (U upper-triangular ones via 4 chained 16x16x4 f32 WMMAs,
// C broadcasts row offsets), then subtract A for the exclusive result.
__global__ __launch_bounds__(32) void scan256_kernel(const uint32_t* __restrict__ binTotals,
                                                     uint32_t* __restrict__ binBase) {
  __shared__ float tot[256];
  __shared__ float rowTot[16];
  __shared__ float rowOff[16];
  int lane = threadIdx.x;
#pragma unroll
  for (int j = 0; j < 8; ++j) tot[j * 32 + lane] = (float)binTotals[j * 32 + lane];
  __syncthreads();
  if (lane < 16) {
    float s = 0.0f;
#pragma unroll
    for (int j = 0; j < 16; ++j) s += tot[lane * 16 + j];
    rowTot[lane] = s;
  }
  __syncthreads();
  if (lane < 16) {
    float s = 0.0f;
    for (int k = 0; k < lane; ++k) s += rowTot[k];
    rowOff[lane] = s;
  }
  __syncthreads();
  int M = lane & 15;
  int hi = lane >> 4;  // 0: lanes 0-15, 1: lanes 16-31
  v8f c;
#pragma unroll
  for (int v = 0; v < 8; ++v) c[v] = rowOff[v + 8 * hi];  // C[i][j] = rowOff[i]
#pragma unroll
  for (int kc = 0; kc < 4; ++kc) {
    int k0 = kc * 4 + (hi ? 2 : 0);
    v2f a, b;
    a.x = tot[M * 16 + k0];
    a.y = tot[M * 16 + k0 + 1];
    b.x = (M >= k0) ? 1.0f : 0.0f;       // U[k][n] = (n >= k)
    b.y = (M >= k0 + 1) ? 1.0f : 0.0f;
    c = __builtin_amdgcn_wmma_f32_16x16x4_f32(false, a, false, b, (short)0, c,
                                              false, false);
  }
#pragma unroll
  for (int v = 0; v < 8; ++v) {
    int p = (v + 8 * hi) * 16 + M;
    binBase[p] = (uint32_t)(c[v] - tot[p]);  // exact integers in f32
  }
}

__global__ __launch_bounds__(256) void scatter_kernel(const uint32_t* __restrict__ srcK,
                                                      const uint32_t* __restrict__ srcV,
                                                      uint32_t* __restrict__ dstK,
                                                      uint32_t* __restrict__ dstV,
                                                      const uint32_t* __restrict__ histExcl,
                                                      const uint32_t* __restrict__ binBase,
                                                      int n, int numTiles, int shift) {
  __shared__ __align__(16) uint32_t ldsK[SORT_TILE];
  __shared__ __align__(16) uint32_t ldsV[SORT_TILE];
  __shared__ uint32_t ldsK2[SORT_TILE];
  __shared__ uint32_t ldsV2[SORT_TILE];
  __shared__ uint32_t cnt[NWAVES][256];
  __shared__ uint32_t sbuf[256];
  __shared__ uint32_t digStart[256];
  __shared__ uint32_t gOff[256];

  int tile = blockIdx.x, t = threadIdx.x;
  int tileStart = tile * SORT_TILE;
  int vcount = n - tileStart; if (vcount > SORT_TILE) vcount = SORT_TILE;
  int wave = t >> 5, lane = t & 31;
  uint32_t lmask = (1u << lane) - 1u;

  for (int j = t; j < NWAVES * 256; j += 256) (&cnt[0][0])[j] = 0u;
  if (t < 32) {
    tdm_load_1d((uint32_t)(uintptr_t)(void*)ldsK, srcK + tileStart,
                SORT_TILE, (uint32_t)(n - tileStart));
    tdm_load_1d((uint32_t)(uintptr_t)(void*)ldsV, srcV + tileStart,
                SORT_TILE, (uint32_t)(n - tileStart));
    __builtin_amdgcn_s_wait_tensorcnt(0);
  }
  __syncthreads();

  // Phase 1: stable rank. Wave w owns linear chunk [w*1024, w*1024+1024).
  uint32_t rankdig[WAVE_ITEMS];
  int base = wave * WAVE_CHUNK;
#pragma unroll
  for (int j = 0; j < WAVE_ITEMS; ++j) {
    int li = base + j * 32 + lane;
    bool valid = li < vcount;
    uint32_t key = ldsK[li];
    uint32_t d = (key >> shift) & 255u;
    unsigned m = (unsigned)__builtin_amdgcn_ballot_w32(valid);
#pragma unroll
    for (int b = 0; b < 8; ++b) {
      unsigned bit = (d >> b) & 1u;
      unsigned vote = (unsigned)__builtin_amdgcn_ballot_w32(bit != 0u);
      m &= bit ? vote : ~vote;
    }
    uint32_t rank = 0u;
    if (valid) {
      uint32_t before = (uint32_t)__popc(m & lmask);
      uint32_t cbase = cnt[wave][d];        // all group lanes read (pre-update)
      if ((m & lmask) == 0u)                // leader (lowest lane of group)
        cnt[wave][d] = cbase + (uint32_t)__popc(m);
      rank = cbase + before;
    }
    rankdig[j] = (d << 16) | rank;          // rank < 1024
  }
  __syncthreads();

  // Per-digit exclusive scan over waves; tile digit totals.
  uint32_t tottd;
  {
    uint32_t s = 0u;
#pragma unroll
    for (int w = 0; w < NWAVES; ++w) { uint32_t c0 = cnt[w][t]; cnt[w][t] = s; s += c0; }
    tottd = s;
  }
  __syncthreads();
  uint32_t dstart = block_scan_excl_256(tottd, sbuf, nullptr);
  digStart[t] = dstart;
  gOff[t] = binBase[t] + histExcl[(size_t)t * (size_t)numTiles + (size_t)tile] - dstart;
  __syncthreads();

  // Phase 2: reorder tile into digit order inside LDS.
#pragma unroll
  for (int j = 0; j < WAVE_ITEMS; ++j) {
    int li = base + j * 32 + lane;
    if (li < vcount) {
      uint32_t rd = rankdig[j];
      uint32_t d = rd >> 16, r = rd & 0xFFFFu;
      uint32_t pos = digStart[d] + cnt[wave][d] + r;
      ldsK2[pos] = ldsK[li];
      ldsV2[pos] = ldsV[li];
    }
  }
  __syncthreads();

  // Phase 3: near-coalesced global scatter (contiguous runs per digit).
  for (int i = t; i < vcount; i += 256) {
    uint32_t k = ldsK2[i];
    uint32_t d = (k >> shift) & 255u;
    uint32_t o = gOff[d] + (uint32_t)i;
    dstK[o] = k;
    dstV[o] = ldsV2[i];
  }
}

// --------------------------- bucket LUT build ------------------------------
__global__ __launch_bounds__(256) void lut_init_kernel(uint32_t* __restrict__ lut,
                                                       int lutN, int n) {
  int i = blockIdx.x * 256 + threadIdx.x;
  if (i < lutN) lut[i] = (uint32_t)n;
}

__global__ __launch_bounds__(256) void lut_build_kernel(const uint32_t* __restrict__ keys,
                                                        uint32_t* __restrict__ lut, int n) {
  int i = blockIdx.x * 256 + threadIdx.x;
  if (i < n) atomicMin(&lut[keys[i] >> LUT_SHIFT], (uint32_t)i);
}

__global__ __launch_bounds__(256) void lut_fill_kernel(uint32_t* __restrict__ lut, int lutN) {
  __shared__ uint32_t s[256];
  int t = threadIdx.x;
  int chunk = (lutN + 255) / 256;
  int lo = t * chunk, hi = lo + chunk; if (hi > lutN) hi = lutN;
  uint32_t m = 0xFFFFFFFFu;
  for (int i = hi - 1; i >= lo; --i) { uint32_t v = lut[i]; m = v < m ? v : m; lut[i] = m; }
  uint32_t acc = m;
  s[t] = acc;
  __syncthreads();
#pragma unroll
  for (int off = 1; off < 256; off <<= 1) {
    uint32_t y = (t + off < 256) ? s[t + off] : 0xFFFFFFFFu;
    __syncthreads();
    acc = y < acc ? y : acc;
    s[t] = acc;
    __syncthreads();
  }
  uint32_t carry = (t + 1 < 256) ? s[t + 1] : 0xFFFFFFFFu;
  for (int i = lo; i < hi; ++i) { uint32_t v = lut[i]; lut[i] = carry < v ? carry : v; }
}

// ------------------------------ interpolation ------------------------------
__device__ __forceinline__ float interp_f(float q, const uint32_t* __restrict__ keys,
                                          const float* __restrict__ vals,
                                          const uint32_t* __restrict__ lut, int n) {
  float x0 = __uint_as_float(keys[0]);
  float xn = __uint_as_float(keys[n - 1]);
  if (q < x0) return vals[0];
  if (q > xn) return vals[n - 1];
  uint32_t b = __float_as_uint(q) >> LUT_SHIFT;  // q in [x0,xn] subset [0,1]
  int lo = (int)lut[b], hi = (int)lut[b + 1];
  while (lo < hi) {                              // searchsorted(side='right')
    int mid = (lo + hi) >> 1;
    if (__uint_as_float(keys[mid]) <= q) lo = mid + 1; else hi = mid;
  }
  int i = lo; if (i < 1) i = 1; if (i > n - 1) i = n - 1;
  float xi = __uint_as_float(keys[i]);
  float xim = __uint_as_float(keys[i - 1]);
  float yi = vals[i];
  float dx = xi - xim;
  if (dx == 0.0f || q == xi) return yi;
  float yim = vals[i - 1];
  return yim + ((q - xim) / dx) * (yi - yim);
}

__global__ __launch_bounds__(256) void interp_kernel(const float* __restrict__ idx,
                                                     const uint32_t* __restrict__ keys,
                                                     const float* __restrict__ vals,
                                                     const uint32_t* __restrict__ lut,
                                                     float* __restrict__ partials, int n) {
  __shared__ float red[256];
  int i = blockIdx.x * 256 + threadIdx.x;
  float g = 0.0f;
  if (i < n - 1) {
    float c0 = fminf(fmaxf(idx[i], 0.0f), 1.0f);
    float c1 = fminf(fmaxf(idx[i + 1], 0.0f), 1.0f);
    float f1 = interp_f(c0 + DELTA_F, keys, vals, lut, n);
    float f2 = interp_f(c1 - DELTA_F, keys, vals, lut, n);
    g = fmaxf(f1 - f2, 0.0f);
  }
  red[threadIdx.x] = g;
  __syncthreads();
#pragma unroll
  for (int off = 128; off > 0; off >>= 1) {
    if (threadIdx.x < off) red[threadIdx.x] += red[threadIdx.x + off];
    __syncthreads();
  }
  if (threadIdx.x == 0) partials[blockIdx.x] = red[0];
}

__global__ __launch_bounds__(256) void final_reduce_kernel(const float* __restrict__ partials,
                                                           int nb, float* __restrict__ out) {
  __shared__ float red[256];
  float s = 0.0f;
  for (int i = threadIdx.x; i < nb; i += 256) s += partials[i];  // fixed order
  red[threadIdx.x] = s;
  __syncthreads();
#pragma unroll
  for (int off = 128; off > 0; off >>= 1) {
    if (threadIdx.x < off) red[threadIdx.x] += red[threadIdx.x + off];
    __syncthreads();
  }
  if (threadIdx.x == 0) out[0] = ALPHA_F * red[0];
}

// ------------------------------- launcher ----------------------------------
extern "C" void kernel_launch(void* const* d_in, const int* in_sizes, int n_in,
                              void* d_out, int out_size, void* d_ws, size_t ws_size,
                              hipStream_t stream) {
  const float* d_indices = (const float*)d_in[0];
  const float* d_array   = (const float*)d_in[1];
  float* out = (float*)d_out;
  int n = in_sizes[0];
  int numTiles = (n + SORT_TILE - 1) / SORT_TILE;

  char* ws = (char*)d_ws;
  size_t nb4 = (size_t)n * 4u;
  uint32_t* keysA = (uint32_t*)(ws);
  uint32_t* valsA = (uint32_t*)(ws + nb4);
  uint32_t* keysB = (uint32_t*)(ws + 2 * nb4);
  uint32_t* valsB = (uint32_t*)(ws + 3 * nb4);
  uint32_t* hist  = (uint32_t*)(ws + 4 * nb4);
  size_t histBytes = (size_t)numTiles * 256u * 4u;
  uint32_t* binTotals = (uint32_t*)(ws + 4 * nb4 + histBytes);
  uint32_t* binBase   = binTotals + 256;
  uint32_t* lut       = binBase + 256;
  float*    partials  = (float*)(lut + LUT_N + 16);

  int nblk = (n + 255) / 256;
  prep_kernel<<<nblk, 256, 0, stream>>>(d_indices, d_array, keysA, valsA, n);

  uint32_t *sk = keysA, *sv = valsA, *dk = keysB, *dv = valsB;
  for (int pass = 0; pass < 4; ++pass) {
    int shift = pass * 8;
    hist_kernel<<<numTiles, 256, 0, stream>>>(sk, hist, n, numTiles, shift);
    scanbins_kernel<<<256, 256, 0, stream>>>(hist, binTotals, numTiles);
    scan256_kernel<<<1, 32, 0, stream>>>(binTotals, binBase);
    scatter_kernel<<<numTiles, 256, 0, stream>>>(sk, sv, dk, dv, hist, binBase,
                                                 n, numTiles, shift);
    uint32_t* tk = sk; sk = dk; dk = tk;
    uint32_t* tv = sv; sv = dv; dv = tv;
  }
  // After 4 passes sorted data is back in keysA/valsA (== sk/sv).

  lut_init_kernel<<<(LUT_N + 255) / 256, 256, 0, stream>>>(lut, LUT_N, n);
  lut_build_kernel<<<nblk, 256, 0, stream>>>(sk, lut, n);
  lut_fill_kernel<<<1, 256, 0, stream>>>(lut, LUT_N);

  int nb = (n - 1 + 255) / 256;
  if (nb > 0) {
    interp_kernel<<<nb, 256, 0, stream>>>(d_indices, sk, (const float*)sv, lut,
                                          partials, n);
  }
  final_reduce_kernel<<<1, 256, 0, stream>>>(partials, nb, out);
}